// NeuroStockBloom_43404939493786
// MI455X (gfx1250) — compile-verified
//
#include <hip/hip_runtime.h>
#include <math.h>

#define NC  617
#define S_  200000
#define T_  15
#define F_  5
#define D_  64
#define A_  768
#define EC_ 20000
#define E_  1000000

typedef float v2f __attribute__((ext_vector_type(2)));
typedef float v8f __attribute__((ext_vector_type(8)));

__device__ __forceinline__ float sigmoidf_(float x){ return 1.0f/(1.0f+expf(-x)); }

// ---------------- BatchNorm over (companies) per timestep ----------------
__global__ void bn_kernel(const float* __restrict__ cts, const float* __restrict__ gamma,
                          const float* __restrict__ beta, float* __restrict__ tsn){
  int t = blockIdx.x;          // 0..14
  int tid = threadIdx.x;       // 256 threads
  __shared__ float ssum[256], ssq[256];
  float s = 0.f, q = 0.f;
  for (int c = tid; c < NC; c += 256){
    float v = cts[c*T_*F_ + t*F_ + (F_-2)];
    s += v; q += v*v;
  }
  ssum[tid] = s; ssq[tid] = q; __syncthreads();
  for (int o = 128; o > 0; o >>= 1){
    if (tid < o){ ssum[tid] += ssum[tid+o]; ssq[tid] += ssq[tid+o]; }
    __syncthreads();
  }
  float mean = ssum[0] / (float)NC;
  float var  = ssq[0] / (float)NC - mean*mean;
  float rstd = rsqrtf(var + 1e-5f);
  float g = gamma[t], b = beta[t];
  for (int c = tid; c < NC; c += 256){
    float v = cts[c*T_*F_ + t*F_ + (F_-2)];
    tsn[c*T_ + t] = (v - mean)*rstd*g + b;
  }
}

// ---------------- 2-layer LSTM + FC + embedding add (per-company block) ----------------
__global__ void lstm_kernel(const float* __restrict__ tsn,
    const float* __restrict__ Wih0, const float* __restrict__ Whh0,
    const float* __restrict__ bih0, const float* __restrict__ bhh0,
    const float* __restrict__ Wih1, const float* __restrict__ Whh1,
    const float* __restrict__ bih1, const float* __restrict__ bhh1,
    const float* __restrict__ fcW, const float* __restrict__ fcb,
    const float* __restrict__ cemb, const int* __restrict__ cids,
    float* __restrict__ compOut)
{
  int c = blockIdx.x;       // company
  int j = threadIdx.x;      // hidden unit, 64 threads
  __shared__ float hbuf[D_];
  __shared__ float hseq[T_*D_];
  float cj = 0.0f;
  hbuf[j] = 0.0f;
  __syncthreads();

  // ---- layer 0 (scalar input) ----
  const float* w0i = Whh0 + (size_t)(0*D_+j)*D_;
  const float* w0f = Whh0 + (size_t)(1*D_+j)*D_;
  const float* w0g = Whh0 + (size_t)(2*D_+j)*D_;
  const float* w0o = Whh0 + (size_t)(3*D_+j)*D_;
  float wi = Wih0[0*D_+j], wf = Wih0[1*D_+j], wg = Wih0[2*D_+j], wo = Wih0[3*D_+j];
  float bi = bih0[0*D_+j]+bhh0[0*D_+j], bf = bih0[1*D_+j]+bhh0[1*D_+j];
  float bg = bih0[2*D_+j]+bhh0[2*D_+j], bo = bih0[3*D_+j]+bhh0[3*D_+j];
  for (int t = 0; t < T_; t++){
    float x = tsn[c*T_ + t];
    float gi = wi*x+bi, gf = wf*x+bf, gg = wg*x+bg, go = wo*x+bo;
    for (int k = 0; k < D_; k++){
      float hk = hbuf[k];
      gi += w0i[k]*hk; gf += w0f[k]*hk; gg += w0g[k]*hk; go += w0o[k]*hk;
    }
    cj = sigmoidf_(gf)*cj + sigmoidf_(gi)*tanhf(gg);
    float hj = sigmoidf_(go)*tanhf(cj);
    __syncthreads();
    hbuf[j] = hj; hseq[t*D_+j] = hj;
    __syncthreads();
  }

  // ---- layer 1 (64-dim input from hseq) ----
  const float* u1i = Wih1 + (size_t)(0*D_+j)*D_;
  const float* u1f = Wih1 + (size_t)(1*D_+j)*D_;
  const float* u1g = Wih1 + (size_t)(2*D_+j)*D_;
  const float* u1o = Wih1 + (size_t)(3*D_+j)*D_;
  const float* w1i = Whh1 + (size_t)(0*D_+j)*D_;
  const float* w1f = Whh1 + (size_t)(1*D_+j)*D_;
  const float* w1g = Whh1 + (size_t)(2*D_+j)*D_;
  const float* w1o = Whh1 + (size_t)(3*D_+j)*D_;
  float bi1 = bih1[0*D_+j]+bhh1[0*D_+j], bf1 = bih1[1*D_+j]+bhh1[1*D_+j];
  float bg1 = bih1[2*D_+j]+bhh1[2*D_+j], bo1 = bih1[3*D_+j]+bhh1[3*D_+j];
  cj = 0.0f;
  __syncthreads();
  hbuf[j] = 0.0f;
  __syncthreads();
  for (int t = 0; t < T_; t++){
    float gi = bi1, gf = bf1, gg = bg1, go = bo1;
    const float* xt = hseq + t*D_;
    for (int k = 0; k < D_; k++){
      float xk = xt[k], hk = hbuf[k];
      gi += u1i[k]*xk + w1i[k]*hk;
      gf += u1f[k]*xk + w1f[k]*hk;
      gg += u1g[k]*xk + w1g[k]*hk;
      go += u1o[k]*xk + w1o[k]*hk;
    }
    cj = sigmoidf_(gf)*cj + sigmoidf_(gi)*tanhf(gg);
    float hj = sigmoidf_(go)*tanhf(cj);
    __syncthreads();
    hbuf[j] = hj;
    __syncthreads();
  }

  // ---- FC + ReLU + company embedding ----
  float s = fcb[j];
  const float* fr = fcW + (size_t)j*D_;
  for (int k = 0; k < D_; k++) s += fr[k]*hbuf[k];
  s = s > 0.0f ? s : 0.0f;
  compOut[(size_t)c*D_ + j] = s + cemb[(size_t)cids[c]*D_ + j];
}

// ---------------- WMMA f32 row-GEMM: out(M,64) = X(M,K) @ Wt(64,K)^T + bias [opt leaky] ----------------
// Block = 256 threads = 8 waves; each wave computes one 16x16 tile; block covers 32 rows x 64 cols.
// Uses V_WMMA_F32_16X16X4_F32 (full fp32 precision). M must be a multiple of 32 (EXEC all-ones).
// Optional second output (out2) duplicates the result (used to seed the c2s accumulator for free).
__global__ void wmma_rowgemm_kernel(const float* __restrict__ X, const float* __restrict__ Wt,
    const float* __restrict__ bias, const float* __restrict__ alphaPtr,
    int K, int leaky, float* __restrict__ out, float* __restrict__ out2)
{
  int wave = threadIdx.x >> 5;
  int lane = threadIdx.x & 31;
  int rowBase = blockIdx.x*32 + (wave >> 2)*16;
  int colBase = (wave & 3)*16;
  int kq  = lane >> 4;     // 0/1: which K-pair this half-wave holds
  int l16 = lane & 15;     // M row (A) / N col (B, C)
  const float* arow = X  + (size_t)(rowBase + l16)*K;
  const float* bcol = Wt + (size_t)(colBase + l16)*K;   // B = Wt^T, so column n = row n of Wt
  v8f acc = {};
  #pragma unroll 8
  for (int k0 = 0; k0 < K; k0 += 4){
    v2f a = *(const v2f*)(arow + k0 + 2*kq);   // A 16x4: VGPR r holds K = 2*(lane/16)+r
    v2f b = *(const v2f*)(bcol + k0 + 2*kq);   // B 4x16: VGPR v holds K = v + 2*(lane/16)
    acc = __builtin_amdgcn_wmma_f32_16x16x4_f32(false, a, false, b, (short)0, acc, false, false);
  }
  float bv = bias[colBase + l16];
  float alpha = leaky ? alphaPtr[0] : 0.0f;
  #pragma unroll
  for (int v = 0; v < 8; v++){
    float h = acc[v] + bv;                      // C/D layout: M = v + 8*(lane/16), N = lane%16
    if (leaky) h = (h >= 0.0f) ? h : alpha*h;
    size_t idx = (size_t)(rowBase + v + 8*kq)*D_ + colBase + l16;
    out[idx] = h;
    if (out2) out2[idx] = h;
  }
}

// ---------------- helpers ----------------
__global__ void copy4_kernel(float4* __restrict__ dst, const float4* __restrict__ src, int n4){
  int i = blockIdx.x*blockDim.x + threadIdx.x;
  if (i < n4) dst[i] = src[i];
}

__global__ void scatter_kernel(float* __restrict__ acc, const float* __restrict__ x,
    const int* __restrict__ src, const int* __restrict__ dst, int nE){
  long long idx = (long long)blockIdx.x*blockDim.x + threadIdx.x;
  if (idx >= (long long)nE*D_) return;
  int e = (int)(idx >> 6);
  int j = (int)(idx & 63);
  unsafeAtomicAdd(&acc[(size_t)dst[e]*D_ + j], x[(size_t)src[e]*D_ + j]);
}

// new_comp = leaky((comp+msg_c2c)@W0^T+b0, a0) + leaky((comp+msg_s2c)@W1^T+b1, a1)
__global__ void comp_transform_kernel(const float* __restrict__ acc0, const float* __restrict__ acc1,
    const float* __restrict__ W0, const float* __restrict__ b0,
    const float* __restrict__ W1, const float* __restrict__ b1,
    const float* __restrict__ ga, int ia0, int ia1, float* __restrict__ out)
{
  int c = blockIdx.x, j = threadIdx.x;   // 64 threads
  __shared__ float x0[D_], x1[D_];
  x0[j] = acc0[(size_t)c*D_ + j];
  x1[j] = acc1[(size_t)c*D_ + j];
  __syncthreads();
  const float* w0 = W0 + (size_t)j*D_;
  const float* w1 = W1 + (size_t)j*D_;
  float s0 = b0[j], s1 = b1[j];
  for (int k = 0; k < D_; k++){ s0 += w0[k]*x0[k]; s1 += w1[k]*x1[k]; }
  float a0 = ga[ia0], a1 = ga[ia1];
  s0 = s0 >= 0.0f ? s0 : a0*s0;
  s1 = s1 >= 0.0f ? s1 : a1*s1;
  out[(size_t)c*D_ + j] = s0 + s1;
}

__global__ void cls_kernel(const float* __restrict__ comp, const float* __restrict__ W,
                           const float* __restrict__ b, float* __restrict__ out){
  int c = blockIdx.x*blockDim.x + threadIdx.x;
  if (c >= NC) return;
  #pragma unroll
  for (int m = 0; m < 2; m++){
    float s = b[m];
    for (int k = 0; k < D_; k++) s += W[m*D_+k]*comp[(size_t)c*D_+k];
    out[c*2 + m] = s;
  }
}

extern "C" void kernel_launch(void* const* d_in, const int* in_sizes, int n_in,
                              void* d_out, int out_size, void* d_ws, size_t ws_size,
                              hipStream_t stream) {
  const float* cts  = (const float*)d_in[0];
  const float* sx   = (const float*)d_in[1];
  const int*   cids = (const int*)d_in[2];
  const int* c2c_s = (const int*)d_in[3];
  const int* c2c_d = (const int*)d_in[4];
  const int* s2c_s = (const int*)d_in[5];
  const int* s2c_d = (const int*)d_in[6];
  const int* c2s_s = (const int*)d_in[7];
  const int* c2s_d = (const int*)d_in[8];
  const float* gamma = (const float*)d_in[9];
  const float* beta  = (const float*)d_in[10];
  const float* Wih0 = (const float*)d_in[11];
  const float* Whh0 = (const float*)d_in[12];
  const float* bih0 = (const float*)d_in[13];
  const float* bhh0 = (const float*)d_in[14];
  const float* Wih1 = (const float*)d_in[15];
  const float* Whh1 = (const float*)d_in[16];
  const float* bih1 = (const float*)d_in[17];
  const float* bhh1 = (const float*)d_in[18];
  const float* fcW  = (const float*)d_in[19];
  const float* fcb  = (const float*)d_in[20];
  const float* cemb = (const float*)d_in[21];
  const float* projW= (const float*)d_in[22];
  const float* projb= (const float*)d_in[23];
  const float* ginW = (const float*)d_in[24];
  const float* ginB = (const float*)d_in[25];
  const float* ginA = (const float*)d_in[26];
  const float* clsW = (const float*)d_in[27];
  const float* clsB = (const float*)d_in[28];

  float* ws = (float*)d_ws;
  size_t off = 0;
  auto alloc = [&](size_t n)->float*{ float* p = ws + off; off += (n + 255) & ~(size_t)255; return p; };
  float* tsn     = alloc((size_t)NC*T_);
  float* compA   = alloc((size_t)NC*D_);
  float* compB   = alloc((size_t)NC*D_);
  float* accC0   = alloc((size_t)NC*D_);
  float* accC1   = alloc((size_t)NC*D_);
  float* sentA   = alloc((size_t)S_*D_);   // sent0, later overwritten by sent1
  float* sentACC = alloc((size_t)S_*D_);   // c2s accumulator (seeded by proj epilogue)

  // 1) BN over companies per timestep
  bn_kernel<<<T_, 256, 0, stream>>>(cts, gamma, beta, tsn);
  // 2) LSTM x2 + FC + embedding -> compA
  lstm_kernel<<<NC, D_, 0, stream>>>(tsn, Wih0, Whh0, bih0, bhh0,
                                     Wih1, Whh1, bih1, bhh1,
                                     fcW, fcb, cemb, cids, compA);
  // 3) sentence projection (big WMMA GEMM, K=768); also seeds sentACC = sent0 for free
  wmma_rowgemm_kernel<<<S_/32, 256, 0, stream>>>(sx, projW, projb, projb, A_, 0, sentA, sentACC);

  float* compCur = compA; float* compNext = compB;
  for (int l = 0; l < 2; l++){
    const bool needSent = (l == 0);   // last layer's new_sent is dead code
    int n4c = NC*D_/4;
    copy4_kernel<<<(n4c+255)/256, 256, 0, stream>>>((float4*)accC0, (const float4*)compCur, n4c);
    copy4_kernel<<<(n4c+255)/256, 256, 0, stream>>>((float4*)accC1, (const float4*)compCur, n4c);
    scatter_kernel<<<(int)(((long long)EC_*D_ + 255)/256), 256, 0, stream>>>(accC0, compCur, c2c_s, c2c_d, EC_);
    scatter_kernel<<<(int)(((long long)E_ *D_ + 255)/256), 256, 0, stream>>>(accC1, sentA,   s2c_s, s2c_d, E_);
    if (needSent)
      scatter_kernel<<<(int)(((long long)E_*D_ + 255)/256), 256, 0, stream>>>(sentACC, compCur, c2s_s, c2s_d, E_);

    const float* W0 = ginW + ((size_t)l*3 + 0)*D_*D_;
    const float* W1 = ginW + ((size_t)l*3 + 1)*D_*D_;
    const float* W2 = ginW + ((size_t)l*3 + 2)*D_*D_;
    const float* b0 = ginB + ((size_t)l*3 + 0)*D_;
    const float* b1 = ginB + ((size_t)l*3 + 1)*D_;
    const float* b2 = ginB + ((size_t)l*3 + 2)*D_;
    comp_transform_kernel<<<NC, D_, 0, stream>>>(accC0, accC1, W0, b0, W1, b1, ginA, l*3+0, l*3+1, compNext);
    if (needSent)  // sent1 = leaky(sentACC @ W2^T + b2); safe to overwrite sentA (s2c already done)
      wmma_rowgemm_kernel<<<S_/32, 256, 0, stream>>>(sentACC, W2, b2, ginA + (l*3+2), D_, 1, sentA, nullptr);

    float* t = compCur; compCur = compNext; compNext = t;
  }
  // 4) classifier
  cls_kernel<<<(NC+63)/64, 64, 0, stream>>>(compCur, clsW, clsB, (float*)d_out);
}